// RegionPooler_64330020160092
// MI455X (gfx1250) — compile-verified
//
#include <hip/hip_runtime.h>
#include <hip/hip_bf16.h>

// Problem constants (from reference)
#define BB 8
#define TT 512
#define PP 2048
#define DD 768

typedef float v2f __attribute__((ext_vector_type(2)));
typedef float v8f __attribute__((ext_vector_type(8)));

// ---------------------------------------------------------------------------
// Kernel 0: global max over token_boxes (non-negative -> uint bit-order works)
// ---------------------------------------------------------------------------
__global__ void k_boxmax(const float* __restrict__ tboxes, unsigned* __restrict__ mx, int n) {
    int i = blockIdx.x * blockDim.x + threadIdx.x;
    if (i < n) atomicMax(mx, __float_as_uint(tboxes[i]));
}

// ---------------------------------------------------------------------------
// Kernel 1: per-patch score = feats·w + b  ->  e = exp(score); patch centers.
// One wave (32 lanes) per (b,p); lanes stride D, shfl_xor reduce.
// ---------------------------------------------------------------------------
__global__ __launch_bounds__(256) void k_scores(const float* __restrict__ feats,
                                                const float* __restrict__ pboxes,
                                                const float* __restrict__ w,
                                                const float* __restrict__ bias,
                                                float* __restrict__ e,
                                                float* __restrict__ cx,
                                                float* __restrict__ cy) {
    int gwid = (blockIdx.x * blockDim.x + threadIdx.x) >> 5;
    int lane = threadIdx.x & 31;
    if (gwid >= BB * PP) return;
    const float* f = feats + (size_t)gwid * DD;
    float s = 0.0f;
    for (int d = lane; d < DD; d += 32) s += f[d] * w[d];
#pragma unroll
    for (int m = 16; m >= 1; m >>= 1) s += __shfl_xor(s, m, 32);
    if (lane == 0) {
        e[gwid] = expf(s + bias[0]);
        const float* pb = pboxes + (size_t)gwid * 4;
        cx[gwid] = 0.5f * (pb[0] + pb[2]);
        cy[gwid] = 0.5f * (pb[1] + pb[3]);
    }
}

// ---------------------------------------------------------------------------
// Kernel 2: per-token S = sum(e over in-box), region_mask, invS (0 if empty).
// One wave per (b,t).
// ---------------------------------------------------------------------------
__global__ __launch_bounds__(256) void k_rowstats(const float* __restrict__ tboxes,
                                                  const unsigned char* __restrict__ tmask,
                                                  const float* __restrict__ e,
                                                  const float* __restrict__ cx,
                                                  const float* __restrict__ cy,
                                                  const unsigned* __restrict__ mxbits,
                                                  float* __restrict__ invS,
                                                  float* __restrict__ maskOut) {
    int gwid = (blockIdx.x * blockDim.x + threadIdx.x) >> 5;
    int lane = threadIdx.x & 31;
    if (gwid >= BB * TT) return;
    int b = gwid / TT;
    float scale = (__uint_as_float(*mxbits) > 1.5f) ? (1.0f / 1000.0f) : 1.0f;
    const float* tb = tboxes + (size_t)gwid * 4;
    float x0 = tb[0] * scale, y0 = tb[1] * scale, x1 = tb[2] * scale, y1 = tb[3] * scale;
    bool tm = tmask[gwid] != 0;
    const float* eb  = e  + (size_t)b * PP;
    const float* cxb = cx + (size_t)b * PP;
    const float* cyb = cy + (size_t)b * PP;
    float s = 0.0f;
    int any = 0;
    for (int p = lane; p < PP; p += 32) {
        float px = cxb[p], py = cyb[p];
        bool in = tm && (px >= x0) && (px <= x1) && (py >= y0) && (py <= y1);
        if (in) { s += eb[p]; any = 1; }
    }
#pragma unroll
    for (int m = 16; m >= 1; m >>= 1) {
        s   += __shfl_xor(s, m, 32);
        any |= __shfl_xor(any, m, 32);
    }
    if (lane == 0) {
        invS[gwid]    = any ? (1.0f / s) : 0.0f;
        maskOut[gwid] = any ? 1.0f : 0.0f;
    }
}

// ---------------------------------------------------------------------------
// Kernel 3: region_feats[t,d] = sum_p (in_box ? e_p*invS_t : 0) * feats[p,d]
// via V_WMMA_F32_16X16X4_F32.  One WG (256 thr = 8 waves) per (b, 16-token
// tile).  Each wave owns 6 N-tiles (96 of D=768 columns).  A-tile (weights)
// built in LDS per 64-patch K-chunk; B frags loaded from global feats.
//
// f32 WMMA lane layouts (ISA 7.12.2):
//   A 16x4:  lane l, vgpr v -> M = l%16, K = (l/16)*2 + v
//   B 4x16:  lane l, vgpr v -> N = l%16, K = (l/16)*2 + v
//   C 16x16: lane l, vgpr r -> M = r + 8*(l/16), N = l%16
// ---------------------------------------------------------------------------
__global__ __launch_bounds__(256) void k_gemm(const float* __restrict__ feats,
                                              const float* __restrict__ tboxes,
                                              const float* __restrict__ e,
                                              const float* __restrict__ cx,
                                              const float* __restrict__ cy,
                                              const float* __restrict__ invS,
                                              const unsigned* __restrict__ mxbits,
                                              float* __restrict__ out) {
    __shared__ float sA[16][65];   // 65 -> conflict-free column reads
    __shared__ float sTB[16][4];
    __shared__ float sInv[16];

    const int b  = blockIdx.x >> 5;          // TT/16 == 32 tiles per batch
    const int t0 = (blockIdx.x & 31) << 4;
    const int tid  = threadIdx.x;
    const int lane = tid & 31;
    const int wid  = tid >> 5;

    if (tid < 16) {
        float scale = (__uint_as_float(*mxbits) > 1.5f) ? (1.0f / 1000.0f) : 1.0f;
        const float* tb = tboxes + ((size_t)b * TT + t0 + tid) * 4;
        sTB[tid][0] = tb[0] * scale;
        sTB[tid][1] = tb[1] * scale;
        sTB[tid][2] = tb[2] * scale;
        sTB[tid][3] = tb[3] * scale;
        sInv[tid]   = invS[b * TT + t0 + tid];
    }
    __syncthreads();

    const float* eb  = e  + (size_t)b * PP;
    const float* cxb = cx + (size_t)b * PP;
    const float* cyb = cy + (size_t)b * PP;
    const float* fbase = feats + (size_t)b * PP * DD;

    const int kk   = tid & 63;        // A-fill: patch within chunk
    const int m4   = tid >> 6;        // A-fill: token sub-index (0..3)
    const int ml   = lane & 15;       // M (for A) / N (for B) lane index
    const int koff = (lane >> 4) << 1;// 0 or 2: K parity-pair per lane half
    const int n0   = wid * 96 + ml;   // first column owned by this lane

    v8f acc[6];
#pragma unroll
    for (int j = 0; j < 6; ++j) acc[j] = (v8f){0, 0, 0, 0, 0, 0, 0, 0};

    for (int kc = 0; kc < PP; kc += 64) {
        // --- build A tile: w[m][kk] = in_box ? e * invS : 0 ---
        {
            float ex = eb[kc + kk], px = cxb[kc + kk], py = cyb[kc + kk];
#pragma unroll
            for (int mi = 0; mi < 4; ++mi) {
                int m = mi * 4 + m4;
                bool in = (px >= sTB[m][0]) && (px <= sTB[m][2]) &&
                          (py >= sTB[m][1]) && (py <= sTB[m][3]);
                sA[m][kk] = in ? ex * sInv[m] : 0.0f;
            }
        }
        __syncthreads();

        // --- 16 K-steps of 4 over this chunk ---
        for (int ks = 0; ks < 64; ks += 4) {
            v2f A;
            A.x = sA[ml][ks + koff];
            A.y = sA[ml][ks + koff + 1];
            const float* fr = fbase + (size_t)(kc + ks + koff) * DD;
#pragma unroll
            for (int j = 0; j < 6; ++j) {
                v2f Bv;
                int n = n0 + j * 16;
                Bv.x = fr[n];
                Bv.y = fr[DD + n];
                acc[j] = __builtin_amdgcn_wmma_f32_16x16x4_f32(
                    false, A, false, Bv, (short)0, acc[j], false, false);
            }
        }
        __syncthreads();
    }

    // --- epilogue: scatter C per ISA layout ---
    const int mbase = (lane >> 4) * 8;
    float* outb = out + ((size_t)b * TT + t0) * DD;
#pragma unroll
    for (int j = 0; j < 6; ++j) {
        int n = n0 + j * 16;
#pragma unroll
        for (int r = 0; r < 8; ++r) {
            outb[(size_t)(mbase + r) * DD + n] = acc[j][r];
        }
    }
}

// ---------------------------------------------------------------------------
extern "C" void kernel_launch(void* const* d_in, const int* in_sizes, int n_in,
                              void* d_out, int out_size, void* d_ws, size_t ws_size,
                              hipStream_t stream) {
    const float* feats  = (const float*)d_in[0];            // (B,P,D)
    const float* tboxes = (const float*)d_in[1];            // (B,T,4)
    const float* pboxes = (const float*)d_in[2];            // (B,P,4)
    const unsigned char* tmask = (const unsigned char*)d_in[3]; // (B,T) bool
    const float* w      = (const float*)d_in[4];            // (D,)
    const float* bias   = (const float*)d_in[5];            // (1,)

    float* out_feats = (float*)d_out;                       // (B,T,D)
    float* out_mask  = (float*)d_out + (size_t)BB * TT * DD;// (B,T)

    // workspace layout (floats)
    float* e    = (float*)d_ws;                 // B*P
    float* cx   = e    + (size_t)BB * PP;       // B*P
    float* cy   = cx   + (size_t)BB * PP;       // B*P
    float* invS = cy   + (size_t)BB * PP;       // B*T
    unsigned* mxbits = (unsigned*)(invS + (size_t)BB * TT); // 1

    hipMemsetAsync(mxbits, 0, sizeof(unsigned), stream);

    {   // box max
        int n = BB * TT * 4;
        k_boxmax<<<(n + 255) / 256, 256, 0, stream>>>(tboxes, mxbits, n);
    }
    {   // scores: one wave per (b,p) -> B*P waves, 8 waves per block
        int blocks = (BB * PP) / 8;
        k_scores<<<blocks, 256, 0, stream>>>(feats, pboxes, w, bias, e, cx, cy);
    }
    {   // row stats: one wave per (b,t)
        int blocks = (BB * TT) / 8;
        k_rowstats<<<blocks, 256, 0, stream>>>(tboxes, tmask, e, cx, cy, mxbits,
                                               invS, out_mask);
    }
    {   // WMMA GEMM: one WG per (b, 16-token tile)
        int blocks = BB * (TT / 16);            // 256 -> one per WGP-ish
        k_gemm<<<blocks, 256, 0, stream>>>(feats, tboxes, e, cx, cy, invS,
                                           mxbits, out_feats);
    }
}